// GCN_15358803051013
// MI455X (gfx1250) — compile-verified
//
#include <hip/hip_runtime.h>

typedef float v2f __attribute__((ext_vector_type(2)));
typedef float v8f __attribute__((ext_vector_type(8)));

#define N_NODES 16384
#define L_IN    200
#define NF      32
#define HC      128
#define NEDGE   524288
#define NGROUP  256
#define NPG     64          // nodes per group
#define SLOPE   0.01f

// ---------------------------------------------------------------------------
// Conv stack: one wave per node, lane = output channel (NF==32==wave32).
// conv1(1->32,k5)+pool5 -> conv2(32->32,k5)+pool5 -> conv3(32->32,k5)+pool3
// Intermediates staged in LDS so every lane can read all channels.
// ---------------------------------------------------------------------------
#define NPB 8   // nodes (waves) per block
__global__ __launch_bounds__(256) void conv_stack_kernel(
    const float* __restrict__ x,
    const float* __restrict__ w1, const float* __restrict__ b1,
    const float* __restrict__ w2, const float* __restrict__ b2,
    const float* __restrict__ w3, const float* __restrict__ b3,
    float* __restrict__ h0)
{
  __shared__ float xs [NPB][L_IN];
  __shared__ float p1s[NPB][NF * 39];
  __shared__ float p2s[NPB][NF * 7];

  const int tid   = threadIdx.x;
  const int node0 = blockIdx.x * NPB;

  for (int i = tid; i < NPB * L_IN; i += 256)
    xs[i / L_IN][i % L_IN] = x[(size_t)(node0 + i / L_IN) * L_IN + (i % L_IN)];
  __syncthreads();

  const int wv = tid >> 5;     // wave = node
  const int c  = tid & 31;     // lane = channel
  const int node = node0 + wv;

  // ---- conv1 + maxpool5: 196 -> 39 ----
  float w1r[5];
#pragma unroll
  for (int j = 0; j < 5; ++j) w1r[j] = w1[c * 5 + j];
  const float b1v = b1[c];
  for (int t = 0; t < 39; ++t) {
    float m = -3.4e38f;
#pragma unroll
    for (int u = 0; u < 5; ++u) {
      const int p = 5 * t + u;
      float y = b1v;
#pragma unroll
      for (int j = 0; j < 5; ++j) y += w1r[j] * xs[wv][p + j];
      m = fmaxf(m, y);
    }
    p1s[wv][c * 39 + t] = m;
  }
  __syncthreads();

  // ---- conv2 + maxpool5: 35 -> 7 ----
  float acc2[35];
#pragma unroll
  for (int t = 0; t < 35; ++t) acc2[t] = 0.f;
  for (int cc = 0; cc < NF; ++cc) {
    float w5[5];
#pragma unroll
    for (int j = 0; j < 5; ++j) w5[j] = w2[(c * NF + cc) * 5 + j];
    float r[39];
#pragma unroll
    for (int t = 0; t < 39; ++t) r[t] = p1s[wv][cc * 39 + t];   // LDS broadcast
#pragma unroll
    for (int t = 0; t < 35; ++t) {
      float s = 0.f;
#pragma unroll
      for (int j = 0; j < 5; ++j) s += w5[j] * r[t + j];
      acc2[t] += s;
    }
  }
  const float b2v = b2[c];
#pragma unroll
  for (int tt = 0; tt < 7; ++tt) {
    float m = -3.4e38f;
#pragma unroll
    for (int u = 0; u < 5; ++u) m = fmaxf(m, acc2[5 * tt + u] + b2v);
    p2s[wv][c * 7 + tt] = m;
  }
  __syncthreads();

  // ---- conv3 + maxpool3: 3 -> 1 ----
  float a3[3] = {0.f, 0.f, 0.f};
  for (int cc = 0; cc < NF; ++cc) {
    float w5[5];
#pragma unroll
    for (int j = 0; j < 5; ++j) w5[j] = w3[(c * NF + cc) * 5 + j];
    float r7[7];
#pragma unroll
    for (int t = 0; t < 7; ++t) r7[t] = p2s[wv][cc * 7 + t];
#pragma unroll
    for (int t = 0; t < 3; ++t) {
#pragma unroll
      for (int j = 0; j < 5; ++j) a3[t] += w5[j] * r7[t + j];
    }
  }
  const float b3v = b3[c];
  h0[(size_t)node * NF + c] = fmaxf(fmaxf(a3[0], a3[1]), a3[2]) + b3v;
}

// ---------------------------------------------------------------------------
// Edge scatter: agg[dst] += h[src].  Thread = (edge, quad of 4 channels).
// float4 load + 4 global f32 atomics (L2-resident working set).
// ---------------------------------------------------------------------------
__global__ __launch_bounds__(256) void scatter_add_kernel(
    const float* __restrict__ h, const int* __restrict__ src,
    const int* __restrict__ dst, float* __restrict__ agg,
    int qshift /* log2(Fin/4) */, int total)
{
  const int tid = blockIdx.x * 256 + threadIdx.x;
  if (tid >= total) return;
  const int e = tid >> qshift;
  const int q = tid & ((1 << qshift) - 1);
  const int fin = 4 << qshift;
  const int s = src[e];
  const int d = dst[e];
  const float4 v = *(const float4*)(h + (size_t)s * fin + q * 4);
  float* ap = agg + (size_t)d * fin + q * 4;
  atomicAdd(ap + 0, v.x);
  atomicAdd(ap + 1, v.y);
  atomicAdd(ap + 2, v.z);
  atomicAdd(ap + 3, v.w);
}

// ---------------------------------------------------------------------------
// Fused dual GEMM + bias + (optional) leaky-ReLU via V_WMMA_F32_16X16X4_F32:
//   out = agg @ Wrel.T + brel + h @ Wroot.T
// Wave owns 16 rows x 128 cols (8 v8f accumulators); K-loop in steps of 4.
// A layout (16x4 f32): lanes 0-15 hold K=k,k+1 ; lanes 16-31 hold K=k+2,k+3.
// C/D layout: VGPR i -> row i (lanes 0-15) / row i+8 (lanes 16-31).
// ---------------------------------------------------------------------------
template <int FIN, int DO_LRELU>
__global__ __launch_bounds__(256) void gemm_dual_kernel(
    const float* __restrict__ agg, const float* __restrict__ h,
    const float* __restrict__ wrel, const float* __restrict__ brel,
    const float* __restrict__ wroot, float* __restrict__ out)
{
  const int lane = threadIdx.x & 31;
  const int wv   = threadIdx.x >> 5;
  const int m0   = (blockIdx.x * 8 + wv) * 16;
  const int lm   = lane & 15;
  const int kh   = lane >> 4;

  v8f acc[8];
#pragma unroll
  for (int nb = 0; nb < 8; ++nb) {
    const float bv = brel[nb * 16 + lm];           // bias per output column
#pragma unroll
    for (int i = 0; i < 8; ++i) acc[nb][i] = bv;
  }

  const float* aptr = agg + (size_t)(m0 + lm) * FIN + 2 * kh;
  const float* hptr = h   + (size_t)(m0 + lm) * FIN + 2 * kh;

  for (int k = 0; k < FIN; k += 4) {
    const v2f a  = *(const v2f*)(aptr + k);
    const v2f ar = *(const v2f*)(hptr + k);
#pragma unroll
    for (int nb = 0; nb < 8; ++nb) {
      const int col = nb * 16 + lm;
      const v2f b  = *(const v2f*)(wrel  + (size_t)col * FIN + k + 2 * kh);
      const v2f br = *(const v2f*)(wroot + (size_t)col * FIN + k + 2 * kh);
      acc[nb] = __builtin_amdgcn_wmma_f32_16x16x4_f32(
          false, a,  false, b,  (short)0, acc[nb], false, false);
      acc[nb] = __builtin_amdgcn_wmma_f32_16x16x4_f32(
          false, ar, false, br, (short)0, acc[nb], false, false);
    }
  }

#pragma unroll
  for (int nb = 0; nb < 8; ++nb) {
    const int col = nb * 16 + lm;
#pragma unroll
    for (int i = 0; i < 8; ++i) {
      float v = acc[nb][i];
      if (DO_LRELU) v = v > 0.f ? v : SLOPE * v;
      out[(size_t)(m0 + i + 8 * kh) * HC + col] = v;
    }
  }
}

// ---------------------------------------------------------------------------
// Group max over 64 contiguous nodes (batch = repeat(arange(G), 64)).
// ---------------------------------------------------------------------------
__global__ __launch_bounds__(128) void segmax_kernel(
    const float* __restrict__ h, float* __restrict__ pooled)
{
  const int g = blockIdx.x;
  const int c = threadIdx.x;
  float m = -3.4e38f;
  const float* p = h + (size_t)g * NPG * HC + c;
  for (int n = 0; n < NPG; ++n) m = fmaxf(m, p[(size_t)n * HC]);
  pooled[(size_t)g * HC + c] = m;
}

// ---------------------------------------------------------------------------
// Final MLP: lrelu(pooled @ lin1.T + b1) @ lin2.T + b2  -> (G, 2)
// ---------------------------------------------------------------------------
__global__ __launch_bounds__(128) void mlp_kernel(
    const float* __restrict__ pooled,
    const float* __restrict__ l1w, const float* __restrict__ l1b,
    const float* __restrict__ l2w, const float* __restrict__ l2b,
    float* __restrict__ out)
{
  __shared__ float row[HC];
  __shared__ float t1[HC];
  const int g = blockIdx.x;
  const int c = threadIdx.x;
  row[c] = pooled[(size_t)g * HC + c];
  __syncthreads();
  float s = l1b[c];
  for (int k = 0; k < HC; ++k) s += l1w[c * HC + k] * row[k];
  t1[c] = s > 0.f ? s : SLOPE * s;
  __syncthreads();
  if (c < 2) {
    float s2 = l2b[c];
    for (int k = 0; k < HC; ++k) s2 += l2w[c * HC + k] * t1[k];
    out[g * 2 + c] = s2;
  }
}

// ---------------------------------------------------------------------------
extern "C" void kernel_launch(void* const* d_in, const int* in_sizes, int n_in,
                              void* d_out, int out_size, void* d_ws, size_t ws_size,
                              hipStream_t stream) {
  const float* x    = (const float*)d_in[0];
  const int*   edge = (const int*)  d_in[1];
  const int*   src  = edge;
  const int*   dst  = edge + NEDGE;
  const float* c1w  = (const float*)d_in[3];
  const float* c1b  = (const float*)d_in[4];
  const float* c2w  = (const float*)d_in[5];
  const float* c2b  = (const float*)d_in[6];
  const float* c3w  = (const float*)d_in[7];
  const float* c3b  = (const float*)d_in[8];
  const float* g1wr = (const float*)d_in[9];
  const float* g1br = (const float*)d_in[10];
  const float* g1wl = (const float*)d_in[11];
  const float* g2wr = (const float*)d_in[12];
  const float* g2br = (const float*)d_in[13];
  const float* g2wl = (const float*)d_in[14];
  const float* g3wr = (const float*)d_in[15];
  const float* g3br = (const float*)d_in[16];
  const float* g3wl = (const float*)d_in[17];
  const float* g4wr = (const float*)d_in[18];
  const float* g4br = (const float*)d_in[19];
  const float* g4wl = (const float*)d_in[20];
  const float* l1w  = (const float*)d_in[21];
  const float* l1b  = (const float*)d_in[22];
  const float* l2w  = (const float*)d_in[23];
  const float* l2b  = (const float*)d_in[24];
  float* out = (float*)d_out;

  char* ws = (char*)d_ws;
  float* h0     = (float*)(ws);                            // N*32   (2 MB)
  float* agg    = (float*)(ws + ((size_t)2  << 20));       // N*128  (8 MB)
  float* hA     = (float*)(ws + ((size_t)10 << 20));       // N*128  (8 MB)
  float* hB     = (float*)(ws + ((size_t)18 << 20));       // N*128  (8 MB)
  float* pooled = (float*)(ws + ((size_t)26 << 20));       // G*128

  // 1) conv stack
  conv_stack_kernel<<<N_NODES / NPB, 256, 0, stream>>>(
      x, c1w, c1b, c2w, c2b, c3w, c3b, h0);

  // 2) graph layer 1 (Fin = 32)
  hipMemsetAsync(agg, 0, (size_t)N_NODES * NF * sizeof(float), stream);
  {
    const int total = NEDGE * (NF / 4);
    scatter_add_kernel<<<(total + 255) / 256, 256, 0, stream>>>(
        h0, src, dst, agg, 3, total);
  }
  gemm_dual_kernel<NF, 1><<<N_NODES / 128, 256, 0, stream>>>(
      agg, h0, g1wr, g1br, g1wl, hA);

  // 3) graph layers 2..4 (Fin = 128)
  const int total128 = NEDGE * (HC / 4);
  hipMemsetAsync(agg, 0, (size_t)N_NODES * HC * sizeof(float), stream);
  scatter_add_kernel<<<(total128 + 255) / 256, 256, 0, stream>>>(
      hA, src, dst, agg, 5, total128);
  gemm_dual_kernel<HC, 1><<<N_NODES / 128, 256, 0, stream>>>(
      agg, hA, g2wr, g2br, g2wl, hB);

  hipMemsetAsync(agg, 0, (size_t)N_NODES * HC * sizeof(float), stream);
  scatter_add_kernel<<<(total128 + 255) / 256, 256, 0, stream>>>(
      hB, src, dst, agg, 5, total128);
  gemm_dual_kernel<HC, 1><<<N_NODES / 128, 256, 0, stream>>>(
      agg, hB, g3wr, g3br, g3wl, hA);

  hipMemsetAsync(agg, 0, (size_t)N_NODES * HC * sizeof(float), stream);
  scatter_add_kernel<<<(total128 + 255) / 256, 256, 0, stream>>>(
      hA, src, dst, agg, 5, total128);
  gemm_dual_kernel<HC, 0><<<N_NODES / 128, 256, 0, stream>>>(
      agg, hA, g4wr, g4br, g4wl, hB);

  // 4) group max-pool
  segmax_kernel<<<NGROUP, HC, 0, stream>>>(hB, pooled);

  // 5) MLP head
  mlp_kernel<<<NGROUP, HC, 0, stream>>>(pooled, l1w, l1b, l2w, l2b, out);
}